// ExpertChoiceRouter_62311385530848
// MI455X (gfx1250) — compile-verified
//
#include <hip/hip_runtime.h>
#include <hip/hip_bf16.h>
#include <math.h>

typedef __attribute__((ext_vector_type(2))) float v2f;
typedef __attribute__((ext_vector_type(8))) float v8f;

#define BB 4
#define SS 4096
#define HH 2048
#define DD 3
#define TOK (BB * SS)          // 16384 tokens
#define TILE_TOK 64            // tokens per workgroup (4 waves x 16)
#define KCHUNK 32              // K consumed per LDS stage
#define LDS_STRIDE 36          // 32 + 4 pad: 144B rows, float4-aligned, conflict-free
#define KSEL 1365              // max(1, S // D)

// ---------------------------------------------------------------------------
// Kernel 1: scores[t][d] = sigmoid( sum_h hs[t][h] * theta[d][h] )
// 16x16 WMMA tiles (f32, K=4), 4 waves per WG, 64-token tile.
// A tile staged global->LDS with CDNA5 async-to-LDS; B tile zero-padded in LDS
// so the WMMA inner loop has no EXEC predication.
// ---------------------------------------------------------------------------
__global__ __launch_bounds__(128) void router_score_kernel(
    const float* __restrict__ hs,      // [TOK][HH]
    const float* __restrict__ theta,   // [DD][HH]
    float* __restrict__ scores)        // [TOK][DD]
{
    __shared__ float smA[TILE_TOK * LDS_STRIDE];   // 9216 B
    __shared__ float smB[16 * LDS_STRIDE];         // 2304 B (rows >= DD are zero)

    const int tid  = threadIdx.x;
    const int wave = tid >> 5;
    const int lane = tid & 31;
    const int m    = lane & 15;    // A row / B row (N) / C column
    const int kh   = lane >> 4;    // K-half selector
    const int rowBase = blockIdx.x * TILE_TOK;     // first token of tile

    const unsigned smA_base = (unsigned)(size_t)&smA[0];   // LDS byte offset

    v8f acc = {};

    for (int kb = 0; kb < HH; kb += KCHUNK) {
        __syncthreads();   // previous chunk fully consumed before overwrite

        // --- A tile: 64 rows x 32 floats, async global->LDS (b128/lane) ----
        #pragma unroll
        for (int i = 0; i < 4; ++i) {
            int f4  = tid + i * 128;      // 0..511, coalesced across lanes
            int r   = f4 >> 3;            // row 0..63 (8 float4 per row)
            int c4  = f4 & 7;             // float4 column
            const float* gp = &hs[(size_t)(rowBase + r) * HH + kb + c4 * 4];
            unsigned ldsb = smA_base + (unsigned)((r * LDS_STRIDE + c4 * 4) * 4);
            asm volatile("global_load_async_to_lds_b128 %0, %1, off"
                         :: "v"(ldsb), "v"(gp) : "memory");
        }

        // --- B tile: 16 x 32 floats, rows >= DD zero-filled ----------------
        {
            int r  = tid >> 3;            // 0..15 (exactly 128 float4 slots)
            int c4 = tid & 7;
            float4 v = {0.0f, 0.0f, 0.0f, 0.0f};
            if (r < DD) v = *(const float4*)&theta[(size_t)r * HH + kb + c4 * 4];
            *(float4*)&smB[r * LDS_STRIDE + c4 * 4] = v;
        }

        // Pull the next chunk toward the caches while this one computes.
        if (kb + KCHUNK < HH) {
            __builtin_prefetch(&hs[(size_t)(rowBase + (tid >> 1)) * HH + kb + KCHUNK], 0, 1);
        }

        asm volatile("s_wait_asynccnt 0x0" ::: "memory");   // own async loads done
        __syncthreads();                                    // all waves staged

        const int arow = wave * 16 + m;   // A row for this lane's M index
        #pragma unroll
        for (int j = 0; j < 8; ++j) {     // 8 x (K=4) WMMAs per 32-wide chunk
            const int ccol = j * 4 + kh * 2;                 // even -> 8B aligned
            v2f a = *(const v2f*)&smA[arow * LDS_STRIDE + ccol];
            v2f b = *(const v2f*)&smB[m    * LDS_STRIDE + ccol];
            acc = __builtin_amdgcn_wmma_f32_16x16x4_f32(
                      false, a, false, b, (short)0, acc, false, false);
        }
    }

    // C/D layout: VGPR r -> lanes 0-15 hold (M=r, N=lane); lanes 16-31 hold (M=8+r).
    if (m < DD) {
        const int tok0 = rowBase + wave * 16 + kh * 8;
        #pragma unroll
        for (int r = 0; r < 8; ++r) {
            float v = acc[r];
            scores[(size_t)(tok0 + r) * DD + m] = 1.0f / (1.0f + __expf(-v));
        }
    }
}

// ---------------------------------------------------------------------------
// Kernel 2: per-batch iterated masked top-k + depth + masks.
// One workgroup per batch row. Exact threshold via binary search on the
// monotonic uint32 encoding of the (positive) sigmoid scores; tie quota keeps
// the selected count at exactly KSEL.
// ---------------------------------------------------------------------------
__global__ __launch_bounds__(256) void router_topk_kernel(
    const float* __restrict__ scores,  // [TOK][DD]
    float* __restrict__ out)           // depth[TOK] | loss[1] | masks[DD][TOK]
{
    const int b   = blockIdx.x;
    const int tid = threadIdx.x;
    const int TPT = SS / 256;          // 16 tokens per thread

    __shared__ unsigned char act[SS];
    __shared__ unsigned char dep[SS];
    __shared__ unsigned      cnt[256];
    __shared__ int           tieTaken;

    for (int i = tid; i < SS; i += 256) { act[i] = 1; dep[i] = 0; }
    __syncthreads();

    for (int r = 0; r < DD; ++r) {
        unsigned       bits[TPT];
        unsigned char  a[TPT];
        #pragma unroll
        for (int i = 0; i < TPT; ++i) {
            int tok = i * 256 + tid;
            a[i]    = act[tok];
            bits[i] = __float_as_uint(scores[((size_t)b * SS + tok) * DD + r]);
        }

        // Largest T with |{active : bits >= T}| >= KSEL  (T = k-th largest).
        unsigned lo = 0u, hi = 0x3F800000u;   // sigmoid in (0,1)
        for (int it = 0; it < 32; ++it) {
            unsigned mid = (unsigned)(((unsigned long long)lo +
                                       (unsigned long long)hi + 1ull) >> 1);
            unsigned c = 0;
            #pragma unroll
            for (int i = 0; i < TPT; ++i) c += (a[i] && bits[i] >= mid) ? 1u : 0u;
            cnt[tid] = c; __syncthreads();
            for (int s = 128; s > 0; s >>= 1) {
                if (tid < s) cnt[tid] += cnt[tid + s];
                __syncthreads();
            }
            unsigned total = cnt[0];
            __syncthreads();
            if (lo < hi) { if (total >= (unsigned)KSEL) lo = mid; else hi = mid - 1; }
        }
        const unsigned T = lo;

        // Strict-above count -> tie quota.
        unsigned c = 0;
        #pragma unroll
        for (int i = 0; i < TPT; ++i) c += (a[i] && bits[i] > T) ? 1u : 0u;
        cnt[tid] = c;
        if (tid == 0) tieTaken = 0;
        __syncthreads();
        for (int s = 128; s > 0; s >>= 1) {
            if (tid < s) cnt[tid] += cnt[tid + s];
            __syncthreads();
        }
        const int quota = KSEL - (int)cnt[0];
        __syncthreads();

        #pragma unroll
        for (int i = 0; i < TPT; ++i) {
            int tok = i * 256 + tid;
            bool sel = false;
            if (a[i]) {
                if (bits[i] > T) sel = true;
                else if (bits[i] == T && atomicAdd(&tieTaken, 1) < quota) sel = true;
            }
            if (sel) dep[tok] = (unsigned char)(r + 1);
            act[tok] = sel ? 1 : 0;     // each token owned by one thread
        }
        __syncthreads();
    }

    float* depth_out = out;
    float* masks_out = out + TOK + 1;   // after depth[TOK] and loss[1]
    for (int i = tid; i < SS; i += 256) {
        int dv = dep[i];
        depth_out[(size_t)b * SS + i] = (float)(dv ? dv : 1);
        float av = act[i] ? 1.0f : 0.0f;
        masks_out[(size_t)b * SS + i]                   = 1.0f; // masks[0] = ones
        masks_out[(size_t)TOK + b * SS + i]             = av;   // masks[1] = final active
        masks_out[(size_t)2 * TOK + (size_t)b * SS + i] = av;   // masks[2] = final active
    }
}

// ---------------------------------------------------------------------------
// Kernel 3: balancing loss (reference double-sigmoids: probs =
// mean(sigmoid(scores)) with scores already post-sigmoid).
// ---------------------------------------------------------------------------
__global__ __launch_bounds__(256) void router_loss_kernel(
    const float* __restrict__ scores,  // [TOK][DD]
    float* __restrict__ out)
{
    __shared__ float r0[256], r1[256], r2[256];
    const int tid = threadIdx.x;
    float s0 = 0.f, s1 = 0.f, s2 = 0.f;
    for (int i = tid; i < TOK; i += 256) {
        const float* p = scores + (size_t)i * DD;
        s0 += 1.0f / (1.0f + __expf(-p[0]));
        s1 += 1.0f / (1.0f + __expf(-p[1]));
        s2 += 1.0f / (1.0f + __expf(-p[2]));
    }
    r0[tid] = s0; r1[tid] = s1; r2[tid] = s2;
    __syncthreads();
    for (int s = 128; s > 0; s >>= 1) {
        if (tid < s) { r0[tid] += r0[tid + s]; r1[tid] += r1[tid + s]; r2[tid] += r2[tid + s]; }
        __syncthreads();
    }
    if (tid == 0) {
        const float inv = 1.0f / (float)TOK;
        const float t   = 1.0f / (float)DD;
        float p0 = r0[0] * inv, p1 = r1[0] * inv, p2 = r2[0] * inv;
        float loss = (t * (logf(t) - logf(p0)) +
                      t * (logf(t) - logf(p1)) +
                      t * (logf(t) - logf(p2))) / (float)DD;
        out[TOK] = loss;
    }
}

// ---------------------------------------------------------------------------
extern "C" void kernel_launch(void* const* d_in, const int* in_sizes, int n_in,
                              void* d_out, int out_size, void* d_ws, size_t ws_size,
                              hipStream_t stream) {
    (void)in_sizes; (void)n_in; (void)out_size; (void)ws_size;
    const float* hs    = (const float*)d_in[0];   // [B,S,H] f32
    const float* theta = (const float*)d_in[1];   // [D,H]   f32
    float* out    = (float*)d_out;                // depth | loss | masks (flat)
    float* scores = (float*)d_ws;                 // [TOK][DD] f32 scratch (192 KB)

    router_score_kernel<<<TOK / TILE_TOK, 128, 0, stream>>>(hs, theta, scores);
    router_topk_kernel<<<BB, 256, 0, stream>>>(scores, out);
    router_loss_kernel<<<1, 256, 0, stream>>>(scores, out);
}